// DGCNN_Seg_89550068121926
// MI455X (gfx1250) — compile-verified
//
#include <hip/hip_runtime.h>
#include <hip/hip_bf16.h>

// ---------------------------------------------------------------------------
// DGCNN forward for MI455X (gfx1250, wave32, WMMA).
// All GEMMs with K>=64 use V_WMMA_F32_16X16X4_F32 (fp32-exact matrix path).
// ---------------------------------------------------------------------------

#define K_NN   40
#define NPTS   2048
#define NB     8
#define NEG_BIG (-3.0e38f)

typedef float v2f __attribute__((ext_vector_type(2)));
typedef float v8f __attribute__((ext_vector_type(8)));

// Only check the builtin on the DEVICE pass; it is (correctly) absent on host.
#if defined(__HIP_DEVICE_COMPILE__)
#if !__has_builtin(__builtin_amdgcn_wmma_f32_16x16x4_f32)
#error "missing __builtin_amdgcn_wmma_f32_16x16x4_f32 on this toolchain (device)"
#endif
#endif

static __device__ __forceinline__ v8f wmma4(v2f a, v2f b, v8f c) {
#if defined(__HIP_DEVICE_COMPILE__)
  // 8 args: (neg_a, A, neg_b, B, c_mod, C, reuse_a, reuse_b)
  return __builtin_amdgcn_wmma_f32_16x16x4_f32(false, a, false, b, (short)0, c,
                                               false, false);
#else
  return c;  // host stub (never executed)
#endif
}

static __device__ __forceinline__ float leaky(float y) {
  return fmaxf(y, 0.2f * y);   // slope < 1 -> branchless leaky relu
}

// ---------------------------------------------------------------------------
// kNN: one block per query row. Distance row in LDS, 40 argmax sweeps with
// lowest-index tie-break (matches lax.top_k; self is included at dist 0).
// ---------------------------------------------------------------------------
template <int C>
__global__ __launch_bounds__(256) void knn_kernel(const float* __restrict__ x,
                                                  int* __restrict__ idx_out) {
  __shared__ float dist[NPTS];
  __shared__ float sctr[C];
  __shared__ float rv[256];
  __shared__ int ri[256];
  const int bn = blockIdx.x;
  const int b = bn >> 11;
  const int n = bn & (NPTS - 1);
  const int tid = threadIdx.x;
  const float* xb = x + (size_t)b * NPTS * C;
  for (int c = tid; c < C; c += 256) sctr[c] = xb[(size_t)n * C + c];
  __syncthreads();
  for (int m = tid; m < NPTS; m += 256) {
    const float* xm = xb + (size_t)m * C;
    float d = 0.f;
#pragma unroll
    for (int c = 0; c < C; ++c) {
      float t = sctr[c] - xm[c];
      d = fmaf(t, t, d);
    }
    dist[m] = -d;
  }
  __syncthreads();
  int* out = idx_out + (size_t)bn * K_NN;
  for (int k = 0; k < K_NN; ++k) {
    float bv = NEG_BIG;
    int bi = NPTS;
    for (int m = tid; m < NPTS; m += 256) {
      float v = dist[m];
      if (v > bv || (v == bv && m < bi)) { bv = v; bi = m; }
    }
    rv[tid] = bv;
    ri[tid] = bi;
    __syncthreads();
    for (int s = 128; s > 0; s >>= 1) {
      if (tid < s) {
        float ov = rv[tid + s];
        int oi = ri[tid + s];
        if (ov > rv[tid] || (ov == rv[tid] && oi < ri[tid])) {
          rv[tid] = ov;
          ri[tid] = oi;
        }
      }
      __syncthreads();
    }
    if (tid == 0) {
      out[k] = ri[0];
      dist[ri[0]] = NEG_BIG;
    }
    __syncthreads();
  }
}

// ---------------------------------------------------------------------------
// Transform stage A: per point (b,n): graph feats (40x6) -> t1 (VALU, 40x64)
// -> t2 via WMMA (48x64 @ 64x128) -> affine/leaky -> max over K -> (128).
// 8 waves, each owns 16 output channels.
// ---------------------------------------------------------------------------
__global__ __launch_bounds__(256) void tstageA_kernel(
    const float* __restrict__ x, const int* __restrict__ idx,
    const float* __restrict__ W1, const float* __restrict__ b1,
    const float* __restrict__ g1, const float* __restrict__ be1,
    const float* __restrict__ W2, const float* __restrict__ b2,
    const float* __restrict__ g2, const float* __restrict__ be2,
    float* __restrict__ h128) {
  __shared__ int sidx[K_NN];
  __shared__ float sctr[3];
  __shared__ float sfeat[K_NN * 6];
  __shared__ float sh1[48 * 66];
  const int bn = blockIdx.x, b = bn >> 11, n = bn & (NPTS - 1);
  const int tid = threadIdx.x;
  const float* xb = x + (size_t)b * NPTS * 3;
  if (tid < K_NN) sidx[tid] = idx[(size_t)bn * K_NN + tid];
  if (tid < 3) sctr[tid] = xb[(size_t)n * 3 + tid];
  __syncthreads();
  if (tid < K_NN) {
    int nb = sidx[tid];
    float cx = sctr[0], cy = sctr[1], cz = sctr[2];
    sfeat[tid * 6 + 0] = xb[(size_t)nb * 3 + 0] - cx;
    sfeat[tid * 6 + 1] = xb[(size_t)nb * 3 + 1] - cy;
    sfeat[tid * 6 + 2] = xb[(size_t)nb * 3 + 2] - cz;
    sfeat[tid * 6 + 3] = cx;
    sfeat[tid * 6 + 4] = cy;
    sfeat[tid * 6 + 5] = cz;
  }
  __syncthreads();
  const float S = rsqrtf(1.0f + 1e-5f);
  for (int o = tid; o < K_NN * 64; o += 256) {
    int k = o >> 6, c = o & 63;
    float acc = 0.f;
#pragma unroll
    for (int j = 0; j < 6; ++j) acc = fmaf(sfeat[k * 6 + j], W1[j * 64 + c], acc);
    float sc = g1[c] * S;
    sh1[k * 66 + c] = leaky(fmaf(sc, acc, fmaf(sc, b1[c], be1[c])));
  }
  for (int o = tid; o < 8 * 64; o += 256) {  // zero pad rows 40..47
    int k = K_NN + (o >> 6), c = o & 63;
    sh1[k * 66 + c] = 0.f;
  }
  __syncthreads();
  const int lane = tid & 31, w = tid >> 5;
  const int col = lane & 15;
  const int colg = w * 16 + col;
  const int khalf = (lane >> 4) * 2;
  const int rhalf = (lane >> 4) * 8;
  const float sc = g2[colg] * S;
  const float bb = fmaf(sc, b2[colg], be2[colg]);
  float vmax = NEG_BIG;
  for (int mt = 0; mt < 3; ++mt) {
    v8f acc = {0.f, 0.f, 0.f, 0.f, 0.f, 0.f, 0.f, 0.f};
    const int row = mt * 16 + col;
    for (int ks = 0; ks < 16; ++ks) {
      const int kb = ks * 4 + khalf;
      v2f a;
      a.x = sh1[row * 66 + kb];
      a.y = sh1[row * 66 + kb + 1];
      v2f bv;
      bv.x = W2[kb * 128 + colg];
      bv.y = W2[(kb + 1) * 128 + colg];
      acc = wmma4(a, bv, acc);
    }
#pragma unroll
    for (int e = 0; e < 8; ++e) {
      int r = mt * 16 + e + rhalf;
      float y = leaky(fmaf(sc, acc[e], bb));
      if (r < K_NN) vmax = fmaxf(vmax, y);
    }
  }
  vmax = fmaxf(vmax, __shfl_xor(vmax, 16, 32));
  if (lane < 16) h128[(size_t)bn * 128 + colg] = vmax;
}

// ---------------------------------------------------------------------------
// Row-tile GEMM (16 rows) -> 1024 channels via WMMA + affine/leaky + per-tile
// row-max partials. CIN=128 (t3, A0=(B,N,128)) or CIN=192 (c6, concat x1..x3).
// 8 waves x 8 Ntiles = 1024 channels. partial: (B,128,1024).
// ---------------------------------------------------------------------------
template <int CIN, int LDA>
__global__ __launch_bounds__(256) void rowgemm_max_kernel(
    const float* __restrict__ A0, const float* __restrict__ A1,
    const float* __restrict__ A2, const float* __restrict__ W,
    const float* __restrict__ bp, const float* __restrict__ gp,
    const float* __restrict__ bep, float* __restrict__ partial) {
  __shared__ float sA[16 * LDA];
  const int blk = blockIdx.x;
  const int b = blk >> 7;
  const int tile = blk & 127;
  const int r0 = tile * 16;
  const int tid = threadIdx.x;
  if constexpr (CIN == 128) {
    for (int o = tid; o < 16 * 128; o += 256) {
      int row = o >> 7, k = o & 127;
      sA[row * LDA + k] = A0[((size_t)(b * NPTS + r0 + row)) * 128 + k];
    }
  } else {
    for (int o = tid; o < 16 * 192; o += 256) {
      int row = o / 192, k = o - row * 192;
      const float* src = (k < 64) ? A0 : ((k < 128) ? A1 : A2);
      sA[row * LDA + k] = src[((size_t)(b * NPTS + r0 + row)) * 64 + (k & 63)];
    }
  }
  __syncthreads();
  const int lane = tid & 31, w = tid >> 5;
  const int col = lane & 15;
  const int khalf = (lane >> 4) * 2;
  const float S = rsqrtf(1.0f + 1e-5f);
  for (int nt = 0; nt < 8; ++nt) {
    const int colg = w * 128 + nt * 16 + col;
    v8f acc = {0.f, 0.f, 0.f, 0.f, 0.f, 0.f, 0.f, 0.f};
    for (int ks = 0; ks < CIN / 4; ++ks) {
      const int kb = ks * 4 + khalf;
      v2f a;
      a.x = sA[col * LDA + kb];
      a.y = sA[col * LDA + kb + 1];
      v2f bv;
      bv.x = W[(size_t)kb * 1024 + colg];
      bv.y = W[(size_t)(kb + 1) * 1024 + colg];
      acc = wmma4(a, bv, acc);
    }
    const float sc = gp[colg] * S;
    const float bb = fmaf(sc, bp[colg], bep[colg]);
    float vmax = NEG_BIG;
#pragma unroll
    for (int e = 0; e < 8; ++e) vmax = fmaxf(vmax, leaky(fmaf(sc, acc[e], bb)));
    vmax = fmaxf(vmax, __shfl_xor(vmax, 16, 32));
    if (lane < 16) partial[((size_t)(b * 128 + tile)) * 1024 + colg] = vmax;
  }
}

// ---------------------------------------------------------------------------
// tl1/tl2/ttr tiny MLP per batch (VALU; ~0.7 MFLOP/block). Reduces the t3
// partials over tiles first. tmat: (B,9).
// ---------------------------------------------------------------------------
__global__ __launch_bounds__(256) void tl_mlp_kernel(
    const float* __restrict__ partial, const float* __restrict__ Wl1,
    const float* __restrict__ bl1, const float* __restrict__ gl1,
    const float* __restrict__ bel1, const float* __restrict__ Wl2,
    const float* __restrict__ bl2, const float* __restrict__ gl2,
    const float* __restrict__ bel2, const float* __restrict__ Wt,
    const float* __restrict__ bt, float* __restrict__ tmat) {
  __shared__ float s0[1024];
  __shared__ float s1[512];
  __shared__ float s2[256];
  const int b = blockIdx.x, tid = threadIdx.x;
  for (int c = tid; c < 1024; c += 256) {
    float m = NEG_BIG;
    for (int t = 0; t < 128; ++t)
      m = fmaxf(m, partial[((size_t)(b * 128 + t)) * 1024 + c]);
    s0[c] = m;
  }
  __syncthreads();
  const float S = rsqrtf(1.0f + 1e-5f);
  for (int o = tid; o < 512; o += 256) {
    float acc = 0.f;
    for (int j = 0; j < 1024; ++j) acc = fmaf(s0[j], Wl1[(size_t)j * 512 + o], acc);
    float sc = gl1[o] * S;
    s1[o] = leaky(fmaf(sc, acc, fmaf(sc, bl1[o], bel1[o])));
  }
  __syncthreads();
  {
    float acc = 0.f;
    for (int j = 0; j < 512; ++j) acc = fmaf(s1[j], Wl2[(size_t)j * 256 + tid], acc);
    float sc = gl2[tid] * S;
    s2[tid] = leaky(fmaf(sc, acc, fmaf(sc, bl2[tid], bel2[tid])));
  }
  __syncthreads();
  if (tid < 9) {
    float acc = bt[tid];
    for (int j = 0; j < 256; ++j) acc = fmaf(s2[j], Wt[j * 9 + tid], acc);
    tmat[b * 9 + tid] = acc;
  }
}

// x' = x @ t   (B,N,3)
__global__ __launch_bounds__(256) void xform_kernel(const float* __restrict__ x,
                                                    const float* __restrict__ tmat,
                                                    float* __restrict__ xT) {
  const int i = blockIdx.x * 256 + threadIdx.x;  // < B*N
  const int b = i >> 11;
  const float* t = tmat + b * 9;
  const float* xi = x + (size_t)i * 3;
  float x0 = xi[0], x1 = xi[1], x2 = xi[2];
  float* o = xT + (size_t)i * 3;
  o[0] = fmaf(x0, t[0], fmaf(x1, t[3], x2 * t[6]));
  o[1] = fmaf(x0, t[1], fmaf(x1, t[4], x2 * t[7]));
  o[2] = fmaf(x0, t[2], fmaf(x1, t[5], x2 * t[8]));
}

// ---------------------------------------------------------------------------
// EdgeConv A (c1): 6 -> 64, max over K. K-dim=6 is too thin for WMMA; VALU.
// ---------------------------------------------------------------------------
__global__ __launch_bounds__(64) void ecA_kernel(
    const float* __restrict__ x, const int* __restrict__ idx,
    const float* __restrict__ W, const float* __restrict__ bp,
    const float* __restrict__ gp, const float* __restrict__ bep,
    float* __restrict__ xout) {
  __shared__ int sidx[K_NN];
  __shared__ float sctr[3];
  __shared__ float sfeat[K_NN * 6];
  const int bn = blockIdx.x, b = bn >> 11, n = bn & (NPTS - 1);
  const int tid = threadIdx.x;
  const float* xb = x + (size_t)b * NPTS * 3;
  if (tid < K_NN) sidx[tid] = idx[(size_t)bn * K_NN + tid];
  if (tid < 3) sctr[tid] = xb[(size_t)n * 3 + tid];
  __syncthreads();
  if (tid < K_NN) {
    int nb = sidx[tid];
    float cx = sctr[0], cy = sctr[1], cz = sctr[2];
    sfeat[tid * 6 + 0] = xb[(size_t)nb * 3 + 0] - cx;
    sfeat[tid * 6 + 1] = xb[(size_t)nb * 3 + 1] - cy;
    sfeat[tid * 6 + 2] = xb[(size_t)nb * 3 + 2] - cz;
    sfeat[tid * 6 + 3] = cx;
    sfeat[tid * 6 + 4] = cy;
    sfeat[tid * 6 + 5] = cz;
  }
  __syncthreads();
  const float S = rsqrtf(1.0f + 1e-5f);
  const int c = tid;
  float wc[6];
#pragma unroll
  for (int j = 0; j < 6; ++j) wc[j] = W[j * 64 + c];
  const float sc = gp[c] * S;
  const float bb = fmaf(sc, bp[c], bep[c]);
  float vmax = NEG_BIG;
  for (int k = 0; k < K_NN; ++k) {
    float acc = 0.f;
#pragma unroll
    for (int j = 0; j < 6; ++j) acc = fmaf(sfeat[k * 6 + j], wc[j], acc);
    vmax = fmaxf(vmax, leaky(fmaf(sc, acc, bb)));
  }
  xout[(size_t)bn * 64 + c] = vmax;
}

// ---------------------------------------------------------------------------
// EdgeConv B (c3/c5): per point, A = 48x128 edge feats (rows 40..47 zero),
// WMMA 48x128 @ 128x64, affine/leaky, masked max over K. 4 waves x 16 cols.
// ---------------------------------------------------------------------------
__global__ __launch_bounds__(128) void ecB_kernel(
    const float* __restrict__ xin, const int* __restrict__ idx,
    const float* __restrict__ W, const float* __restrict__ bp,
    const float* __restrict__ gp, const float* __restrict__ bep,
    float* __restrict__ xout) {
  __shared__ float sA[48 * 130];
  __shared__ int sidx[K_NN];
  __shared__ float sctr[64];
  const int bn = blockIdx.x, b = bn >> 11, n = bn & (NPTS - 1);
  const int tid = threadIdx.x;
  const float* xb = xin + (size_t)b * NPTS * 64;
  if (tid < K_NN) sidx[tid] = idx[(size_t)bn * K_NN + tid];
  for (int c = tid; c < 64; c += 128) sctr[c] = xb[(size_t)n * 64 + c];
  __syncthreads();
  for (int o = tid; o < 48 * 128; o += 128) {
    int row = o >> 7, k = o & 127;
    float v = 0.f;
    if (row < K_NN)
      v = (k < 64) ? (xb[(size_t)sidx[row] * 64 + k] - sctr[k]) : sctr[k - 64];
    sA[row * 130 + k] = v;
  }
  __syncthreads();
  const int lane = tid & 31, w = tid >> 5;
  const int col = lane & 15;
  const int colg = w * 16 + col;
  const int khalf = (lane >> 4) * 2;
  const int rhalf = (lane >> 4) * 8;
  const float S = rsqrtf(1.0f + 1e-5f);
  const float sc = gp[colg] * S;
  const float bb = fmaf(sc, bp[colg], bep[colg]);
  float vmax = NEG_BIG;
  for (int mt = 0; mt < 3; ++mt) {
    v8f acc = {0.f, 0.f, 0.f, 0.f, 0.f, 0.f, 0.f, 0.f};
    const int row = mt * 16 + col;
    for (int ks = 0; ks < 32; ++ks) {
      const int kb = ks * 4 + khalf;
      v2f a;
      a.x = sA[row * 130 + kb];
      a.y = sA[row * 130 + kb + 1];
      v2f bv;
      bv.x = W[kb * 64 + colg];
      bv.y = W[(kb + 1) * 64 + colg];
      acc = wmma4(a, bv, acc);
    }
#pragma unroll
    for (int e = 0; e < 8; ++e) {
      int r = mt * 16 + e + rhalf;
      float y = leaky(fmaf(sc, acc[e], bb));
      if (r < K_NN) vmax = fmaxf(vmax, y);
    }
  }
  vmax = fmaxf(vmax, __shfl_xor(vmax, 16, 32));
  if (lane < 16) xout[(size_t)bn * 64 + colg] = vmax;
}

// out[b,c] = max over 128 tile-partials
__global__ __launch_bounds__(256) void finalmax_kernel(
    const float* __restrict__ partial, float* __restrict__ out) {
  const int i = blockIdx.x * 256 + threadIdx.x;  // < 8192
  const int b = i >> 10, c = i & 1023;
  float m = NEG_BIG;
  for (int t = 0; t < 128; ++t)
    m = fmaxf(m, partial[((size_t)(b * 128 + t)) * 1024 + c]);
  out[i] = m;
}

// ---------------------------------------------------------------------------
// Host side
// ---------------------------------------------------------------------------
extern "C" void kernel_launch(void* const* d_in, const int* in_sizes, int n_in,
                              void* d_out, int out_size, void* d_ws,
                              size_t ws_size, hipStream_t stream) {
  (void)in_sizes; (void)n_in; (void)out_size; (void)ws_size;
  // Inputs in setup_inputs() insertion order:
  // 0:x | t1:1-4 | t2:5-8 | t3:9-12 | tl1:13-16 | tl2:17-20 | ttr_W:21
  // ttr_b:22 | c1:23-26 | c3:27-30 | c5:31-34 | c6:35-38  (each mk: W,b,g,be)
  const float* x = (const float*)d_in[0];
  auto F = [&](int i) { return (const float*)d_in[i]; };

  char* wsb = (char*)d_ws;
  // ws layout (bytes), total ~28.0 MB:
  int*   idx     = (int*)(wsb + 0);            // 8*2048*40*4   = 2621440
  float* h128    = (float*)(wsb + 2621440);    // 8*2048*128*4  = 8388608
  float* partial = (float*)(wsb + 11010048);   // 8*128*1024*4  = 4194304
  float* tmat    = (float*)(wsb + 15204352);   // 8*9*4 (pad 512)
  float* xT      = (float*)(wsb + 15204864);   // 8*2048*3*4    = 196608
  float* x1      = (float*)(wsb + 15401472);   // 8*2048*64*4   = 4194304
  float* x2      = (float*)(wsb + 19595776);   // 4194304
  float* x3      = (float*)(wsb + 23790080);   // 4194304 -> end 27984384

  const int NPT = NB * NPTS;  // 16384 point-blocks

  // ---- transform branch ----
  knn_kernel<3><<<NPT, 256, 0, stream>>>(x, idx);
  tstageA_kernel<<<NPT, 256, 0, stream>>>(x, idx, F(1), F(2), F(3), F(4), F(5),
                                          F(6), F(7), F(8), h128);
  rowgemm_max_kernel<128, 130><<<NB * 128, 256, 0, stream>>>(
      h128, nullptr, nullptr, F(9), F(10), F(11), F(12), partial);
  tl_mlp_kernel<<<NB, 256, 0, stream>>>(partial, F(13), F(14), F(15), F(16),
                                        F(17), F(18), F(19), F(20), F(21),
                                        F(22), tmat);
  xform_kernel<<<NPT / 256, 256, 0, stream>>>(x, tmat, xT);

  // ---- edge-conv stack ----
  knn_kernel<3><<<NPT, 256, 0, stream>>>(xT, idx);
  ecA_kernel<<<NPT, 64, 0, stream>>>(xT, idx, F(23), F(24), F(25), F(26), x1);
  knn_kernel<64><<<NPT, 256, 0, stream>>>(x1, idx);
  ecB_kernel<<<NPT, 128, 0, stream>>>(x1, idx, F(27), F(28), F(29), F(30), x2);
  knn_kernel<64><<<NPT, 256, 0, stream>>>(x2, idx);
  ecB_kernel<<<NPT, 128, 0, stream>>>(x2, idx, F(31), F(32), F(33), F(34), x3);

  // ---- c6 + global max ----
  rowgemm_max_kernel<192, 194><<<NB * 128, 256, 0, stream>>>(
      x1, x2, x3, F(35), F(36), F(37), F(38), partial);
  finalmax_kernel<<<32, 256, 0, stream>>>(partial, (float*)d_out);
}